// GraphCNN_22239340658851
// MI455X (gfx1250) — compile-verified
//
#include <hip/hip_runtime.h>
#include <math.h>

typedef __attribute__((ext_vector_type(2))) float v2f;
typedef __attribute__((ext_vector_type(8))) float v8f;

#define FEAT 128
#define NSTEPS 4

// ---------------- utility kernels ----------------

__global__ void zero_u32(unsigned* __restrict__ p, int n) {
    int i = blockIdx.x * blockDim.x + threadIdx.x;
    if (i < n) p[i] = 0u;
}

// initial total += column sum of x  (blockDim = 128, block handles 256 nodes)
__global__ void colsum_kernel(const float* __restrict__ x, float* __restrict__ out, int N) {
    int d = threadIdx.x;
    int n0 = blockIdx.x * 256;
    float s = 0.f;
    for (int i = 0; i < 256; ++i) {
        int n = n0 + i;
        if (n < N) s += x[(size_t)n * FEAT + d];
    }
    atomicAdd(&out[d], s);
}

__global__ void degree_kernel(const int* __restrict__ row, int* __restrict__ deg, int E) {
    int i = blockIdx.x * blockDim.x + threadIdx.x;
    int stride = gridDim.x * blockDim.x;
    for (int e = i; e < E; e += stride) atomicAdd(&deg[row[e]], 1);
}

// single-workgroup exclusive scan: deg[N] -> rowptr[N+1], cursor[N]
__global__ void scan_kernel(const int* __restrict__ deg, int* __restrict__ rowptr,
                            int* __restrict__ cursor, int N) {
    __shared__ int sd[1024];
    __shared__ int carry;
    if (threadIdx.x == 0) carry = 0;
    __syncthreads();
    for (int base = 0; base < N; base += 1024) {
        int i = base + threadIdx.x;
        int v = (i < N) ? deg[i] : 0;
        sd[threadIdx.x] = v;
        __syncthreads();
        for (int off = 1; off < 1024; off <<= 1) {
            int t = (threadIdx.x >= (unsigned)off) ? sd[threadIdx.x - off] : 0;
            __syncthreads();
            sd[threadIdx.x] += t;
            __syncthreads();
        }
        int excl = sd[threadIdx.x] - v;
        if (i < N) { rowptr[i] = carry + excl; cursor[i] = carry + excl; }
        __syncthreads();
        if (threadIdx.x == 1023) carry += sd[1023];
        __syncthreads();
    }
    if (threadIdx.x == 0) rowptr[N] = carry;
}

__global__ void fill_kernel(const int* __restrict__ row, const int* __restrict__ col,
                            int* __restrict__ cursor, int* __restrict__ colidx, int E) {
    int i = blockIdx.x * blockDim.x + threadIdx.x;
    int stride = gridDim.x * blockDim.x;
    for (int e = i; e < E; e += stride) {
        int r = row[e];
        int p = atomicAdd(&cursor[r], 1);
        colidx[p] = col[e];
    }
}

// Precompute DFT cos/sin matrices directly in WMMA-B-fragment layout:
// frag index gid = (w*32 + t)*32 + lane ; element (K = 4t + 2*(lane/16) + {0,1}, Ncol = 16w + lane%16)
__global__ void dft_frag_kernel(float* __restrict__ Cfrag, float* __restrict__ Sfrag) {
    int gid = blockIdx.x * blockDim.x + threadIdx.x;
    if (gid >= 8 * 32 * 32) return;
    int lane = gid & 31;
    int t    = (gid >> 5) & 31;
    int w    = gid >> 10;
    int k    = 4 * t + 2 * (lane >> 4);
    int colj = 16 * w + (lane & 15);
    const float step = 6.283185307179586f / 128.0f;
    float a0 = step * (float)((k * colj) & 127);
    float a1 = step * (float)(((k + 1) * colj) & 127);
    Cfrag[2 * gid + 0] = cosf(a0);
    Cfrag[2 * gid + 1] = cosf(a1);
    Sfrag[2 * gid + 0] = sinf(a0);
    Sfrag[2 * gid + 1] = sinf(a1);
}

// ---------------- SpMM (gather-mean with fused roll) ----------------
// one wave per node; lane holds 4 features (float4); colidx fetched cooperatively
__global__ void spmm_kernel(const float* __restrict__ h, const int* __restrict__ rowptr,
                            const int* __restrict__ colidx, float* __restrict__ pooled, int N) {
    int wave = (blockIdx.x * blockDim.x + threadIdx.x) >> 5;
    int lane = threadIdx.x & 31;
    if (wave >= N) return;
    int s = rowptr[wave], e = rowptr[wave + 1];
    float ax = 0.f, ay = 0.f, az = 0.f, aw = 0.f;
    for (int base = s; base < e; base += 32) {
        int cnt = e - base;
        if (cnt > 32) cnt = 32;
        // one coalesced load covers the next 32 edges for the whole wave
        int myc = (base + lane < e) ? colidx[base + lane] : 0;
        for (int j = 0; j < cnt; ++j) {
            int c = __shfl(myc, j, 32);
            const float4 v = *(const float4*)(h + (size_t)c * FEAT + lane * 4);
            ax += v.x; ay += v.y; az += v.z; aw += v.w;
        }
    }
    float cntf = (float)(e - s);
    float inv = cntf > 0.f ? 1.f / cntf : 0.f;
    int d0 = lane * 4;
    float* pr = pooled + (size_t)wave * FEAT;
    // pooled[d] = mean_c h[c][d-1]  -> store accumulated d' at position (d'+1) mod 128
    pr[(d0 + 1) & 127] = ax * inv;
    pr[(d0 + 2) & 127] = ay * inv;
    pr[(d0 + 3) & 127] = az * inv;
    pr[(d0 + 4) & 127] = aw * inv;
}

// ---------------- bind (circular conv via fp32 DFT GEMMs on WMMA) ----------------
// block = 256 threads = 8 waves; block handles 16 nodes; wave w owns freq/feature cols [16w,16w+16)
__global__ __launch_bounds__(256)
void bind_kernel(float* __restrict__ h, const float* __restrict__ pooled,
                 const float* __restrict__ Cfrag, const float* __restrict__ Sfrag,
                 float* __restrict__ total, int N) {
    __shared__ float hs[16][132];
    __shared__ float ps[16][132];
    __shared__ float zr[16][132];
    __shared__ float zi[16][132];

    int tid = threadIdx.x;
    int node0 = blockIdx.x * 16;

    // async cooperative load of the 16x128 h and pooled tiles straight into LDS
    // (ISA: INST_OFFSET is added to both the LDS and the global address)
    {
        int r  = tid >> 4;          // 0..15
        int cb = (tid & 15) * 8;    // 0..120
        const float* hg = h + (size_t)(node0 + r) * FEAT + cb;
        const float* pg = pooled + (size_t)(node0 + r) * FEAT + cb;
        unsigned hs0 = (unsigned)(size_t)&hs[r][cb];   // low 32 bits of flat addr = LDS offset
        unsigned ps0 = (unsigned)(size_t)&ps[r][cb];
        asm volatile("global_load_async_to_lds_b128 %0, %1, off"           :: "v"(hs0), "v"(hg) : "memory");
        asm volatile("global_load_async_to_lds_b128 %0, %1, off offset:16" :: "v"(hs0), "v"(hg) : "memory");
        asm volatile("global_load_async_to_lds_b128 %0, %1, off"           :: "v"(ps0), "v"(pg) : "memory");
        asm volatile("global_load_async_to_lds_b128 %0, %1, off offset:16" :: "v"(ps0), "v"(pg) : "memory");
        asm volatile("s_wait_asynccnt 0x0" ::: "memory");
    }
    __syncthreads();

    int w    = tid >> 5;
    int lane = tid & 31;
    int ln   = lane & 15;
    int lh   = lane >> 4;

    // DFT B-fragments for this wave's 16 columns, kept in VGPRs for both stages
    v2f cf[32], sf[32];
    {
        const v2f* cfp = (const v2f*)Cfrag + (size_t)w * 32 * 32 + lane;
        const v2f* sfp = (const v2f*)Sfrag + (size_t)w * 32 * 32 + lane;
#pragma unroll
        for (int t = 0; t < 32; ++t) { cf[t] = cfp[t * 32]; sf[t] = sfp[t * 32]; }
    }

    // Stage 1: A = h@C, B = h@S, U = p@C, V = p@S  (contraction over K=128, k-tiles of 4)
    v8f accA = {}, accB = {}, accU = {}, accV = {};
#pragma unroll
    for (int t = 0; t < 32; ++t) {
        int kk = 4 * t + 2 * lh;
        v2f ah; ah.x = hs[ln][kk]; ah.y = hs[ln][kk + 1];
        v2f ap; ap.x = ps[ln][kk]; ap.y = ps[ln][kk + 1];
        accA = __builtin_amdgcn_wmma_f32_16x16x4_f32(false, ah, false, cf[t], (short)0, accA, false, false);
        accB = __builtin_amdgcn_wmma_f32_16x16x4_f32(false, ah, false, sf[t], (short)0, accB, false, false);
        accU = __builtin_amdgcn_wmma_f32_16x16x4_f32(false, ap, false, cf[t], (short)0, accU, false, false);
        accV = __builtin_amdgcn_wmma_f32_16x16x4_f32(false, ap, false, sf[t], (short)0, accV, false, false);
    }

    // elementwise complex product (per node, per frequency): Zre = AU-BV, Zim = AV+BU
    {
        int jcol = w * 16 + ln;
#pragma unroll
        for (int v = 0; v < 8; ++v) {
            float a = accA[v], b = accB[v], u = accU[v], q = accV[v];
            int m = v + 8 * lh;
            zr[m][jcol] = a * u - b * q;
            zi[m][jcol] = a * q + b * u;
        }
    }
    __syncthreads();

    // Stage 2: out = (Zre@C + Zim@S) / 128
    v8f accO = {};
#pragma unroll
    for (int t = 0; t < 32; ++t) {
        int kk = 4 * t + 2 * lh;
        v2f azr; azr.x = zr[ln][kk]; azr.y = zr[ln][kk + 1];
        v2f azi; azi.x = zi[ln][kk]; azi.y = zi[ln][kk + 1];
        accO = __builtin_amdgcn_wmma_f32_16x16x4_f32(false, azr, false, cf[t], (short)0, accO, false, false);
        accO = __builtin_amdgcn_wmma_f32_16x16x4_f32(false, azi, false, sf[t], (short)0, accO, false, false);
    }

    // h_new = sign(conv/128 + h); write in place; accumulate column sums
    int dcol = w * 16 + ln;
    float csum = 0.f;
#pragma unroll
    for (int v = 0; v < 8; ++v) {
        int m = v + 8 * lh;
        float val = accO[v] * (1.0f / 128.0f) + hs[m][dcol];
        float hn = (val > 0.f) ? 1.f : ((val < 0.f) ? -1.f : 0.f);
        h[(size_t)(node0 + m) * FEAT + dcol] = hn;
        csum += hn;
    }
    csum += __shfl_xor(csum, 16, 32);
    if (lane < 16) atomicAdd(&total[dcol], csum);
}

// ---------------- launcher ----------------

extern "C" void kernel_launch(void* const* d_in, const int* in_sizes, int n_in,
                              void* d_out, int out_size, void* d_ws, size_t ws_size,
                              hipStream_t stream) {
    const float* x  = (const float*)d_in[0];
    const int*   ei = (const int*)d_in[1];
    int N = in_sizes[0] / FEAT;
    int E = in_sizes[1] / 2;
    const int* row = ei;
    const int* col = ei + E;

    char* ws = (char*)d_ws;
    size_t off = 0;
    auto alloc = [&](size_t bytes) -> void* {
        void* p = ws + off;
        off = (off + bytes + 255) & ~((size_t)255);
        return p;
    };
    float* h      = (float*)alloc((size_t)N * FEAT * sizeof(float));
    float* pooled = (float*)alloc((size_t)N * FEAT * sizeof(float));
    int*   colidx = (int*)alloc((size_t)E * sizeof(int));
    int*   deg    = (int*)alloc((size_t)N * sizeof(int));
    int*   rowptr = (int*)alloc((size_t)(N + 1) * sizeof(int));
    int*   cursor = (int*)alloc((size_t)N * sizeof(int));
    float* Cfrag  = (float*)alloc((size_t)8 * 32 * 32 * 2 * sizeof(float));
    float* Sfrag  = (float*)alloc((size_t)8 * 32 * 32 * 2 * sizeof(float));
    (void)ws_size; (void)n_in; (void)out_size;

    float* total = (float*)d_out;

    // init
    zero_u32<<<1, 128, 0, stream>>>((unsigned*)total, FEAT);
    zero_u32<<<(N + 255) / 256, 256, 0, stream>>>((unsigned*)deg, N);
    colsum_kernel<<<(N + 255) / 256, 128, 0, stream>>>(x, total, N);
    hipMemcpyAsync(h, x, (size_t)N * FEAT * sizeof(float), hipMemcpyDeviceToDevice, stream);

    // CSR build (once; reused by all 4 layers)
    degree_kernel<<<1024, 256, 0, stream>>>(row, deg, E);
    scan_kernel<<<1, 1024, 0, stream>>>(deg, rowptr, cursor, N);
    fill_kernel<<<1024, 256, 0, stream>>>(row, col, cursor, colidx, E);
    dft_frag_kernel<<<(8 * 32 * 32 + 255) / 256, 256, 0, stream>>>(Cfrag, Sfrag);

    // 4 propagation layers
    for (int layer = 0; layer < NSTEPS; ++layer) {
        spmm_kernel<<<(N * 32 + 255) / 256, 256, 0, stream>>>(h, rowptr, colidx, pooled, N);
        bind_kernel<<<N / 16, 256, 0, stream>>>(h, pooled, Cfrag, Sfrag, total, N);
    }
}